// GaussianPointCloudRasterisation_9285719294482
// MI455X (gfx1250) — compile-verified
//
#include <hip/hip_runtime.h>
#include <hip/hip_bf16.h>

typedef __attribute__((ext_vector_type(2))) float v2f;
typedef __attribute__((ext_vector_type(8))) float v8f;

#define TILE_SZ 16
#define PPT     64
#define TU_N    32
#define NT_N    1024
#define HW      512
#define WSTR    17   // padded stride for w LDS (bank-conflict-free)

__global__ __launch_bounds__(256)
void gs_raster_kernel(const float* __restrict__ point_uv,
                      const float* __restrict__ point_conic,
                      const float* __restrict__ point_alpha,
                      const float* __restrict__ point_color,
                      const float* __restrict__ point_depth,
                      const int*   __restrict__ tile_point_indices,
                      float* __restrict__ out_img,
                      float* __restrict__ out_depth,
                      float* __restrict__ out_alpha,
                      float* __restrict__ out_lastoff,
                      float* __restrict__ out_vcount)
{
    __shared__ float ptu[PPT], ptv[PPT], pca[PPT], pcb[PPT], pcc[PPT], pal[PPT];
    __shared__ float Bm[PPT * 16];       // B matrix: 64 x 16 (cols 0..4 = r,g,b,depth,1)
    __shared__ float Wm[256 * WSTR];     // one K=16 chunk of W (256 pixels x 16 k), padded
    __shared__ float norm_lds[256];

    const int tid = threadIdx.x;
    const int t   = blockIdx.x;
    const int tu  = t % TU_N;
    const int tv  = t / TU_N;

    // ---- preload per-tile point data + build B ----
    if (tid < PPT) {
        const int idx = tile_point_indices[t * PPT + tid];
        ptu[tid] = point_uv[idx * 2 + 0];
        ptv[tid] = point_uv[idx * 2 + 1];
        pca[tid] = point_conic[idx * 3 + 0];
        pcb[tid] = point_conic[idx * 3 + 1];
        pcc[tid] = point_conic[idx * 3 + 2];
        pal[tid] = point_alpha[idx];
        float* Bk = &Bm[tid * 16];
        Bk[0] = point_color[idx * 3 + 0];
        Bk[1] = point_color[idx * 3 + 1];
        Bk[2] = point_color[idx * 3 + 2];
        Bk[3] = point_depth[idx];
        Bk[4] = 1.0f;
#pragma unroll
        for (int c = 5; c < 16; ++c) Bk[c] = 0.0f;
    }
    __syncthreads();

    // ---- per-pixel scan state (pixel == tid) ----
    const int   pu = tid & 15, pv = tid >> 4;
    const float px = (float)(tu * TILE_SZ + pu) + 0.5f;
    const float py = (float)(tv * TILE_SZ + pv) + 0.5f;

    float T = 1.0f, Tfin = 1.0f, norm = 0.0f;
    int   vc = 0, last = 0;
    bool  sat = false;

    const int wave  = tid >> 5;
    const int lane  = tid & 31;
    const int m16   = lane & 15;          // A row / B col / D col index
    const int khalf = (lane >> 4) << 1;   // 0 or 2: K offset for lanes 16-31

    v8f acc[2] = {};                       // two 16x16 D tiles per wave

    for (int chunk = 0; chunk < 4; ++chunk) {
        // --- phase 1: transmittance scan for K = chunk*16 .. +15 ---
#pragma unroll
        for (int kl = 0; kl < 16; ++kl) {
            const int k = chunk * 16 + kl;
            const float dx    = px - ptu[k];
            const float dy    = py - ptv[k];
            const float power = -0.5f * (pca[k] * dx * dx + pcc[k] * dy * dy)
                                - pcb[k] * dx * dy;
            const float alpha = __expf(power) * pal[k];
            const bool  vis     = alpha >= (1.0f / 255.0f);
            const float alpha_c = vis ? fminf(alpha, 0.99f) : 0.0f;
            const float one_m   = 1.0f - alpha_c;
            const bool  trig    = vis && (T * one_m < 1e-4f);
            const bool  satn    = sat || trig;
            const bool  contrib = vis && !satn;
            const float w       = contrib ? alpha_c * T : 0.0f;
            Wm[tid * WSTR + kl] = w;
            norm += w;
            if (contrib) { Tfin *= one_m; vc += 1; last = k + 1; }
            T   *= one_m;   // Ti is the unconditional exclusive cumprod
            sat  = satn;
        }
        __syncthreads();

        // --- phase 2: WMMA accumulate  D(16x16) += W_tile(16x4) x B_tile(4x16) ---
#pragma unroll
        for (int t2 = 0; t2 < 2; ++t2) {
            const int row = (wave * 2 + t2) * 16 + m16;   // pixel row for A fragment
#pragma unroll
            for (int kc = 0; kc < 4; ++kc) {
                const int klA = kc * 4 + khalf;           // k within chunk for this lane
                const int kg  = chunk * 16 + klA;         // global k for B
                v2f a, b;
                a.x = Wm[row * WSTR + klA];
                a.y = Wm[row * WSTR + klA + 1];
                b.x = Bm[kg * 16 + m16];
                b.y = Bm[(kg + 1) * 16 + m16];
                acc[t2] = __builtin_amdgcn_wmma_f32_16x16x4_f32(
                    false, a, false, b, (short)0, acc[t2], false, false);
            }
        }
        __syncthreads();
    }

    // ---- per-pixel scalar outputs + norm handoff ----
    norm_lds[tid] = norm;
    {
        const int gx = tu * TILE_SZ + pu;
        const int gy = tv * TILE_SZ + pv;
        const int gp = gy * HW + gx;
        out_alpha[gp]   = 1.0f - Tfin;
        out_lastoff[gp] = (float)(t * PPT + last);
        out_vcount[gp]  = (float)vc;
    }
    __syncthreads();

    // ---- write WMMA results: D layout -> image / depth ----
#pragma unroll
    for (int t2 = 0; t2 < 2; ++t2) {
        const v8f c = acc[t2];
#pragma unroll
        for (int r = 0; r < 8; ++r) {
            const int mloc  = r + ((lane & 16) ? 8 : 0);
            const int pixel = (wave * 2 + t2) * 16 + mloc;
            const int ppu = pixel & 15, ppv = pixel >> 4;
            const int gx = tu * TILE_SZ + ppu;
            const int gy = tv * TILE_SZ + ppv;
            if (m16 < 3) {
                out_img[(gy * HW + gx) * 3 + m16] = c[r];
            } else if (m16 == 3) {
                const float nm = norm_lds[pixel];
                out_depth[gy * HW + gx] = c[r] / fmaxf(nm, 1e-6f);
            }
        }
    }
}

extern "C" void kernel_launch(void* const* d_in, const int* in_sizes, int n_in,
                              void* d_out, int out_size, void* d_ws, size_t ws_size,
                              hipStream_t stream) {
    const float* point_uv    = (const float*)d_in[0];
    const float* point_conic = (const float*)d_in[1];
    const float* point_alpha = (const float*)d_in[2];
    const float* point_color = (const float*)d_in[3];
    const float* point_depth = (const float*)d_in[4];
    const int*   tpi         = (const int*)d_in[5];

    float* out         = (float*)d_out;
    float* out_img     = out;                 // 512*512*3 = 786432
    float* out_depth   = out + 786432;        // 512*512
    float* out_alpha   = out + 1048576;       // 512*512
    float* out_lastoff = out + 1310720;       // 512*512
    float* out_vcount  = out + 1572864;       // 512*512

    gs_raster_kernel<<<NT_N, 256, 0, stream>>>(
        point_uv, point_conic, point_alpha, point_color, point_depth, tpi,
        out_img, out_depth, out_alpha, out_lastoff, out_vcount);
}